// NSDLikeSurfaceLoss_40286793236654
// MI455X (gfx1250) — compile-verified
//
#include <hip/hip_runtime.h>
#include <stdint.h>

#define Hdim 192
#define Wdim 192
#define NSLICE 24              // B*C = 8*3
#define BR 32                  // output rows per workgroup
#define NBLK (Hdim / BR)       // 6 row-blocks per slice
#define RADIUS 16              // window: sigmoid(4*(3-d)) underflows for d>16
#define HALO 17                // RADIUS + 1 (erosion neighborhood)
#define TR (BR + 2 * HALO)     // 66 raw target rows staged in LDS
#define BNDR (BR + 2 * RADIUS) // 64 boundary rows
#define BND_WPR (Wdim / 32)    // 6 bitmask words per row
#define GPITCH (Wdim + 2 * RADIUS) // 224: padded g8 row
#define NTHREADS 256
#define NWAVES (NTHREADS / 32)
#define LUTN 514               // band LUT: d2 in [0,512], 513 = "far" -> 0

// ---- LDS layout (bytes) ----
#define RAW_OFF 0
#define RAW_BYTES (TR * Wdim * 4)              // 50688
#define G8_OFF (RAW_OFF + RAW_BYTES)           // 50688
#define G8_BYTES (BR * GPITCH)                 // 7168
#define MB_OFF (G8_OFF + G8_BYTES)             // 57856
#define MB_BYTES (TR * BND_WPR * 4)            // 1584
#define BND_OFF (MB_OFF + MB_BYTES)            // 59440
#define BND_BYTES (BNDR * BND_WPR * 4)         // 1536
#define CB_OFF (BND_OFF + BND_BYTES)           // 60976: column bits, 2 words/col
#define CB_BYTES (Wdim * 8)                    // 1536
#define RED_OFF (CB_OFF + CB_BYTES)            // 62512
#define RED_BYTES 64
#define LUT_OFF (RED_OFF + RED_BYTES)          // 62576
#define LUT_BYTES (LUTN * 4)                   // 2056
#define SMEM_BYTES (LUT_OFF + LUT_BYTES)       // 64632

typedef __attribute__((ext_vector_type(2))) float v2f;
typedef __attribute__((ext_vector_type(8))) float v8f;

__global__ __launch_bounds__(NTHREADS) void nsd_band_kernel(
    const float* __restrict__ probs, const int* __restrict__ target,
    float* __restrict__ partials) {
  __shared__ __align__(16) unsigned char smem[SMEM_BYTES];
  int* raw = (int*)(smem + RAW_OFF);             // [TR][Wdim]
  unsigned char* g8 = smem + G8_OFF;             // [BR][GPITCH], padded
  unsigned* mbits = (unsigned*)(smem + MB_OFF);  // [TR][BND_WPR]
  unsigned* bnd = (unsigned*)(smem + BND_OFF);   // [BNDR][BND_WPR]
  unsigned* cb = (unsigned*)(smem + CB_OFF);     // [Wdim][2] column bits
  float* red = (float*)(smem + RED_OFF);
  float* lut = (float*)(smem + LUT_OFF);

  const int tid = threadIdx.x;
  const int lane = tid & 31;
  const int wv = tid >> 5;
  const int wg = blockIdx.x;
  const int slice = wg / NBLK;
  const int rb = wg % NBLK;
  const int y0 = rb * BR;
  const int ytop = y0 - HALO;  // global y of raw row 0

  // ---- Phase 1: async DMA valid target rows into LDS (ASYNCcnt path) ----
  const int yv0 = (ytop > 0) ? ytop : 0;
  const int yv1 = (y0 + BR + HALO < Hdim) ? (y0 + BR + HALO) : Hdim;
  {
    const long long src_elem =
        (long long)slice * (Hdim * Wdim) + (long long)yv0 * Wdim;
    const unsigned long long gsrc =
        (unsigned long long)(uintptr_t)(target + src_elem);
    const unsigned lbase = (unsigned)(uintptr_t)smem + RAW_OFF +
                           (unsigned)(yv0 - ytop) * (Wdim * 4);
    const int n16 = (yv1 - yv0) * (Wdim * 4 / 16);  // 16B chunks, aligned
    for (int i = tid; i < n16; i += NTHREADS) {
      unsigned long long ga = gsrc + (unsigned long long)i * 16ull;
      unsigned la = lbase + (unsigned)i * 16u;
      asm volatile("global_load_async_to_lds_b128 %0, %1, off" ::"v"(la),
                   "v"(ga)
                   : "memory");
    }
  }

  // ---- Fill band LUT while the DMA is in flight ----
  for (int i = tid; i < LUTN; i += NTHREADS) {
    float b = 0.0f;
    if (i <= 512) {
      float d = sqrtf((float)i);
      b = 1.0f / (1.0f + expf(4.0f * (d - 3.0f)));
    }
    lut[i] = b;
  }

  asm volatile("s_wait_asynccnt 0" ::: "memory");
  __syncthreads();

  // ---- Phase 2a: ballot-pack gt mask into bit words (0 outside image) ----
  for (int w = wv; w < TR * BND_WPR; w += NWAVES) {
    const int row = w / BND_WPR;  // raw tile row
    const int y = ytop + row;
    const int valid = ((unsigned)y < (unsigned)Hdim) ? 1 : 0;
    const int v = valid ? (raw[w * 32 + lane] != 0) : 0;
    unsigned long long bw = __ballot(v);
    if (lane == 0) mbits[w] = (unsigned)bw;
  }
  __syncthreads();

  // ---- Phase 2b: word-parallel boundary = c & ~(n & s & w & e) ----
  for (int w = tid; w < BNDR * BND_WPR; w += NTHREADS) {
    const int r = w / BND_WPR;  // boundary row r <-> y = y0-16+r
    const int xw = w % BND_WPR;
    const int mrow = r + 1;     // matching mask row in mbits
    const unsigned c = mbits[mrow * BND_WPR + xw];
    const unsigned n = mbits[(mrow - 1) * BND_WPR + xw];
    const unsigned s = mbits[(mrow + 1) * BND_WPR + xw];
    const unsigned wcar = (xw > 0) ? (mbits[mrow * BND_WPR + xw - 1] >> 31) : 0u;
    const unsigned ecar =
        (xw < BND_WPR - 1) ? (mbits[mrow * BND_WPR + xw + 1] << 31) : 0u;
    const unsigned wm = (c << 1) | wcar;
    const unsigned em = (c >> 1) | ecar;
    bnd[w] = c & ~(n & s & wm & em);
  }
  __syncthreads();

  // ---- Phase 2c: ballot-transpose boundary bits into per-column words ----
  for (int q = wv; q < Wdim * 2; q += NWAVES) {
    const int x = q >> 1;
    const int half = q & 1;
    const int r = half * 32 + lane;  // boundary row
    const unsigned bv = (bnd[r * BND_WPR + (x >> 5)] >> (x & 31)) & 1u;
    unsigned long long bw = __ballot((int)bv);
    if (lane == 0) cb[x * 2 + half] = (unsigned)bw;
  }
  __syncthreads();

  // ---- Phase 3: column pass via clz/ctz on the 33-bit row window ----
  // pad columns of g8 with 255
  for (int i = tid; i < BR * 2 * RADIUS; i += NTHREADS) {
    const int yo = i / (2 * RADIUS);
    const int j = i % (2 * RADIUS);
    const int c = (j < RADIUS) ? j : (Wdim + j);  // [0,16) U [208,224)
    g8[yo * GPITCH + c] = (unsigned char)255;
  }
  for (int i = tid; i < BR * Wdim; i += NTHREADS) {
    const int yo = i / Wdim;
    const int x = i % Wdim;
    const unsigned long long col =
        ((unsigned long long)cb[x * 2 + 1] << 32) | cb[x * 2];
    // bit k of m: boundary at dy = k - 16 (rows yo..yo+32)
    const unsigned long long m = (col >> yo) & 0x1FFFFFFFFull;
    const unsigned lo = (unsigned)m & 0x1FFFFu;          // dy <= 0
    const unsigned hi = (unsigned)(m >> 16) & 0x1FFFFu;  // dy >= 0
    const int gneg = lo ? (__builtin_clz(lo) - 15) : 255;
    const int gpos = hi ? __builtin_ctz(hi) : 255;
    const int g = gneg < gpos ? gneg : gpos;
    g8[yo * GPITCH + RADIUS + x] = (unsigned char)g;
  }
  __syncthreads();

  // ---- Phase 4: branchless row pass + LUT band + masked accumulation ----
  float num = 0.f, den = 0.f;
  for (int i = tid; i < BR * Wdim; i += NTHREADS) {
    const int yo = i / Wdim;
    const int x = i % Wdim;
    const unsigned char* grow = g8 + yo * GPITCH + x;  // grow[t] <-> dx = t-16
    int d2 = 0x7FFFFFFF;
#pragma unroll
    for (int t = 0; t <= 2 * RADIUS; ++t) {
      const int gv = grow[t];
      const int dx = t - RADIUS;
      const int cand = gv * gv + dx * dx;
      d2 = d2 < cand ? d2 : cand;
    }
    const int idx = d2 > 512 ? 513 : d2;  // "far" -> band 0
    const float band = lut[idx];
    const float tt = (raw[(yo + HALO) * Wdim + x] != 0) ? 1.0f : 0.0f;
    const float p = probs[(long long)slice * (Hdim * Wdim) +
                          (long long)(y0 + yo) * Wdim + x];
    const float bt = band * tt;
    num += p * bt;
    den += bt;
  }

  // ---- Phase 5: exact-f32 wave reduction via V_WMMA_F32_16X16X4_F32 ----
  // B (4x16 f32): vgpr0 = num (rows k=0,2), vgpr1 = den (rows k=1,3).
  // A_num keeps even-k rows, A_den odd-k rows: C[i][j] = lane_j + lane_{j+16}.
  {
    v2f an;
    an[0] = 1.0f;
    an[1] = 0.0f;
    v2f ad;
    ad[0] = 0.0f;
    ad[1] = 1.0f;
    v2f bb;
    bb[0] = num;
    bb[1] = den;
    v8f zc = {};
    v8f cn = __builtin_amdgcn_wmma_f32_16x16x4_f32(false, an, false, bb,
                                                   (short)0, zc, false, false);
    v8f cd = __builtin_amdgcn_wmma_f32_16x16x4_f32(false, ad, false, bb,
                                                   (short)0, zc, false, false);
    float sn = cn[0];
    float sd = cd[0];
#pragma unroll
    for (int mask = 8; mask >= 1; mask >>= 1) {  // fold remaining 16 columns
      sn += __shfl_xor(sn, mask, 32);
      sd += __shfl_xor(sd, mask, 32);
    }
    if (lane == 0) {
      red[wv * 2 + 0] = sn;
      red[wv * 2 + 1] = sd;
    }
  }
  __syncthreads();
  if (tid == 0) {
    float tn = 0.f, td = 0.f;
    for (int w = 0; w < NWAVES; ++w) {
      tn += red[w * 2 + 0];
      td += red[w * 2 + 1];
    }
    partials[wg * 2 + 0] = tn;  // fixed slot: deterministic, no atomics
    partials[wg * 2 + 1] = td;
  }
}

__global__ __launch_bounds__(32) void nsd_finalize_kernel(
    const float* __restrict__ partials, float* __restrict__ out, int nwg) {
  const int lane = threadIdx.x;
  float n = 0.f, d = 0.f;
  for (int i = lane; i < nwg; i += 32) {
    n += partials[2 * i + 0];
    d += partials[2 * i + 1];
  }
#pragma unroll
  for (int mask = 16; mask >= 1; mask >>= 1) {
    n += __shfl_xor(n, mask, 32);
    d += __shfl_xor(d, mask, 32);
  }
  if (lane == 0) out[0] = 1.0f - n / fmaxf(d, 1.0f);
}

extern "C" void kernel_launch(void* const* d_in, const int* in_sizes, int n_in,
                              void* d_out, int out_size, void* d_ws,
                              size_t ws_size, hipStream_t stream) {
  const float* probs = (const float*)d_in[0];
  const int* target = (const int*)d_in[1];
  float* partials = (float*)d_ws;  // 144*2 floats, fully rewritten every call
  const int nwg = NSLICE * NBLK;   // 144 workgroups
  nsd_band_kernel<<<nwg, NTHREADS, 0, stream>>>(probs, target, partials);
  nsd_finalize_kernel<<<1, 32, 0, stream>>>(partials, (float*)d_out, nwg);
}